// QuantumBottle_46351287058607
// MI455X (gfx1250) — compile-verified
//
#include <hip/hip_runtime.h>
#include <hip/hip_bf16.h>

typedef float v2f __attribute__((ext_vector_type(2)));
typedef float v4f __attribute__((ext_vector_type(4)));
typedef float v8f __attribute__((ext_vector_type(8)));

// ---------------------------------------------------------------------------
// Setup: build U^T (16x16 complex, row-major) and M2 = S @ W^T (16x16, cols>=4
// are zero) from q_weights / W.  One wave; lane t owns column t of U.
// ---------------------------------------------------------------------------
__global__ void qb_setup(const float* __restrict__ qw, const float* __restrict__ W,
                         float* __restrict__ bu_r, float* __restrict__ bu_i,
                         float* __restrict__ m2) {
    int t = threadIdx.x;
    if (t >= 16) return;
    float ur[16], ui[16];
#pragma unroll
    for (int r = 0; r < 16; ++r) { ur[r] = (r == t) ? 1.f : 0.f; ui[r] = 0.f; }

    for (int l = 0; l < 2; ++l) {
        // shared Rot gates: Rot(phi,theta,omega) = RZ(omega) RY(theta) RZ(phi)
        for (int w = 0; w < 4; ++w) {
            float phi = qw[(l * 4 + w) * 3 + 0];
            float th  = qw[(l * 4 + w) * 3 + 1];
            float om  = qw[(l * 4 + w) * 3 + 2];
            float ct, st, sap, cap, sam, cam;
            __sincosf(0.5f * th, &st, &ct);
            __sincosf(0.5f * (phi + om), &sap, &cap);
            __sincosf(0.5f * (phi - om), &sam, &cam);
            // ep = cap - i sap ; em = cam - i sam
            float g00r =  cap * ct, g00i = -sap * ct;   // ep*ct
            float g01r = -cam * st, g01i = -sam * st;   // -conj(em)*st
            float g10r =  cam * st, g10i = -sam * st;   // em*st
            float g11r =  cap * ct, g11i =  sap * ct;   // conj(ep)*ct
            int msk = 1 << (3 - w);
#pragma unroll
            for (int r0 = 0; r0 < 16; ++r0) {
                if (r0 & msk) continue;
                int r1 = r0 | msk;
                float xr = ur[r0], xi = ui[r0], yr = ur[r1], yi = ui[r1];
                ur[r0] = g00r * xr - g00i * xi + g01r * yr - g01i * yi;
                ui[r0] = g00r * xi + g00i * xr + g01r * yi + g01i * yr;
                ur[r1] = g10r * xr - g10i * xi + g11r * yr - g11i * yi;
                ui[r1] = g10r * xi + g10i * xr + g11r * yi + g11i * yr;
            }
        }
        // CNOT ring: control w, target (w+1)%4  (wire w lives at bit 3-w)
        for (int w = 0; w < 4; ++w) {
            int cm = 1 << (3 - w);
            int tm = 1 << (3 - ((w + 1) & 3));
#pragma unroll
            for (int r0 = 0; r0 < 16; ++r0) {
                if (!(r0 & cm) || (r0 & tm)) continue;
                int r1 = r0 | tm;
                float tr = ur[r0], ti = ui[r0];
                ur[r0] = ur[r1]; ui[r0] = ui[r1];
                ur[r1] = tr;     ui[r1] = ti;
            }
        }
    }
    // lane t holds U[:,t]  ==  row t of U^T
#pragma unroll
    for (int r = 0; r < 16; ++r) {
        bu_r[t * 16 + r] = ur[r];
        bu_i[t * 16 + r] = ui[r];
    }
    // M2[k][n] = sum_w sign_w(k) * W[n][w]   (n<4), sign_w(k) = 1-2*bit(k,3-w)
    for (int n = 0; n < 16; ++n) {
        float v = 0.f;
        if (n < 4) {
#pragma unroll
            for (int w = 0; w < 4; ++w) {
                float s = ((t >> (3 - w)) & 1) ? -1.f : 1.f;
                v += s * W[n * 4 + w];
            }
        }
        m2[t * 16 + n] = v;
    }
}

// ---------------------------------------------------------------------------
// Main: one 16-sample tile per wave per iteration.
//   GEMM1: D1 = Amp(16x16) @ U^T  (Re and Im)  -> 8 x v_wmma_f32_16x16x4_f32
//   P = Dr^2 + Di^2 ; LDS transpose D-layout -> A-layout
//   GEMM2: Out = P @ M2 + b(C-operand)        -> 4 x v_wmma_f32_16x16x4_f32
//   Output tile bounced through LDS -> one coalesced b128 NT store per lane.
// ---------------------------------------------------------------------------
__global__ __launch_bounds__(256) void qb_main(const float* __restrict__ z,
                                               const float* __restrict__ scale_p,
                                               const float* __restrict__ bvec,
                                               const float* __restrict__ bu_r,
                                               const float* __restrict__ bu_i,
                                               const float* __restrict__ m2,
                                               float* __restrict__ out,
                                               int ntiles) {
    __shared__ float lds[8 * 256];           // 1 KB per wave, private region
    const int lane = threadIdx.x & 31;
    const int wid  = threadIdx.x >> 5;
    const int g    = lane >> 4;              // lane half: K/row-group select
    const int nn   = lane & 15;              // B/C/D column index
    float* tlds = &lds[wid * 256];

    // Persistent B operands: lane holds B[k = 4c+2g+j][n = nn]
    v2f bR[4], bI[4], bM[4];
#pragma unroll
    for (int c = 0; c < 4; ++c) {
#pragma unroll
        for (int j = 0; j < 2; ++j) {
            int k = 4 * c + 2 * g + j;
            bR[c][j] = bu_r[k * 16 + nn];
            bI[c][j] = bu_i[k * 16 + nn];
            bM[c][j] = m2 [k * 16 + nn];
        }
    }
    const float cinit = (nn < 4) ? bvec[nn] : 0.0f;
    const float hs = 0.5f * scale_p[0];

    const int wave   = blockIdx.x * (blockDim.x >> 5) + wid;
    const int nwaves = gridDim.x * (blockDim.x >> 5);

    for (int tile = wave; tile < ntiles; tile += nwaves) {
        if (tile + nwaves < ntiles)
            __builtin_prefetch(z + (size_t)((tile + nwaves) * 16 + nn) * 4, 0, 3);

        // per-sample RY product-state factors (lane's A-row sample)
        const v4f zr = *(const v4f*)(z + (size_t)(tile * 16 + nn) * 4);
        float c0, s0, c1, s1, c2, s2, c3, s3;
        __sincosf(tanhf(zr.x) * hs, &s0, &c0);
        __sincosf(tanhf(zr.y) * hs, &s1, &c1);
        __sincosf(tanhf(zr.z) * hs, &s2, &c2);
        __sincosf(tanhf(zr.w) * hs, &s3, &c3);
        const float w2f = g ? s2 : c2;       // bit1 of k is fixed = g

        v8f dR = {}, dI = {};
#pragma unroll
        for (int c = 0; c < 4; ++c) {
            // k = 4c + 2g + j : bit3=c>>1 (wire0), bit2=c&1 (wire1),
            //                   bit1=g (wire2),    bit0=j (wire3)
            float base = ((c & 2) ? s0 : c0) * ((c & 1) ? s1 : c1) * w2f;
            v2f a; a[0] = base * c3; a[1] = base * s3;
            dR = __builtin_amdgcn_wmma_f32_16x16x4_f32(false, a, false, bR[c],
                                                       (short)0, dR, false, false);
            dI = __builtin_amdgcn_wmma_f32_16x16x4_f32(false, a, false, bI[c],
                                                       (short)0, dI, false, false);
        }

        // probs in D layout -> LDS (row-major P[m][n]) -> reload in A layout
#pragma unroll
        for (int v = 0; v < 8; ++v) {
            float p = dR[v] * dR[v] + dI[v] * dI[v];
            tlds[(v + 8 * g) * 16 + nn] = p;
        }
        asm volatile("s_wait_dscnt 0" ::: "memory");   // wave-local RAW order

        v8f d2 = {cinit, cinit, cinit, cinit, cinit, cinit, cinit, cinit};
#pragma unroll
        for (int c = 0; c < 4; ++c) {
            v2f pA;
            pA[0] = tlds[nn * 16 + 4 * c + 2 * g + 0];
            pA[1] = tlds[nn * 16 + 4 * c + 2 * g + 1];
            d2 = __builtin_amdgcn_wmma_f32_16x16x4_f32(false, pA, false, bM[c],
                                                       (short)0, d2, false, false);
        }

        // Out tile (16 samples x 4) -> LDS row-major, then one coalesced
        // 16B nontemporal store per lane (lanes 0..15 cover the 256B tile).
        if (nn < 4) {
#pragma unroll
            for (int v = 0; v < 8; ++v)
                tlds[(v + 8 * g) * 4 + nn] = d2[v];
        }
        asm volatile("s_wait_dscnt 0" ::: "memory");
        if (lane < 16) {
            v4f o = *(const v4f*)&tlds[lane * 4];
            __builtin_nontemporal_store(o, (v4f*)(out + (size_t)tile * 64 + lane * 4));
        }
    }
}

extern "C" void kernel_launch(void* const* d_in, const int* in_sizes, int n_in,
                              void* d_out, int out_size, void* d_ws, size_t ws_size,
                              hipStream_t stream) {
    const float* z     = (const float*)d_in[0];
    const float* scale = (const float*)d_in[1];
    const float* qw    = (const float*)d_in[2];
    const float* W     = (const float*)d_in[3];
    const float* bvec  = (const float*)d_in[4];
    float* out = (float*)d_out;

    float* bu_r = (float*)d_ws;          // 256 floats
    float* bu_i = bu_r + 256;            // 256 floats
    float* m2   = bu_i + 256;            // 256 floats

    const int B      = in_sizes[0] / 4;
    const int ntiles = B / 16;

    qb_setup<<<1, 32, 0, stream>>>(qw, W, bu_r, bu_i, m2);
    qb_main<<<1024, 256, 0, stream>>>(z, scale, bvec, bu_r, bu_i, m2, out, ntiles);
}